// GraphAttention_16054587752771
// MI455X (gfx1250) — compile-verified
//
#include <hip/hip_runtime.h>
#include <hip/hip_bf16.h>
#include <math.h>

typedef float v2f __attribute__((ext_vector_type(2)));
typedef float v8f __attribute__((ext_vector_type(8)));

#define BSZ   16
#define GSZ   1024
#define DD    128
#define HH    8
#define FF    16
#define MTOT  (BSZ*GSZ)          // 16384 rows
#define ALPHA 0.2f
#define BN_EPS 1e-5f
#define NEGV  (-9.0e15f)

#define LDA_P 132                // padded LDS row stride (floats); 528B, 16B-aligned

// ---------------------------------------------------------------------------
// GEMM: C[M,128] = A[M,128] @ W[128,128]^T  (+bias, optional ReLU)
// via V_WMMA_F32_16X16X4_F32.
// Block = 256 threads (8 waves) owns a 64x128 tile of C:
//   - A tile (64x128, 32KB) staged in LDS once with coalesced float4 loads
//   - each wave owns one 16-wide N tile; A fragments come from LDS (ds_load_b64,
//     bank-conflict-free via +4 float row pad), B fragments direct from global.
// ---------------------------------------------------------------------------
__global__ void __launch_bounds__(256)
gemm_nt_wmma(const float* __restrict__ A, const float* __restrict__ W,
             const float* __restrict__ bias, float* __restrict__ C,
             int M, int do_relu)
{
    __shared__ float As[64 * LDA_P];             // 33.8 KB

    const int K    = 128;
    const int tid  = threadIdx.x;
    const int row0 = blockIdx.x * 64;

    // ---- cooperative A-tile stage: 64 rows x 32 float4 = 2048 float4 ----
#pragma unroll
    for (int it = 0; it < 8; ++it) {
        int idx = it * 256 + tid;                // 0..2047
        int r   = idx >> 5;                      // row 0..63
        int c4  = idx & 31;                      // float4 column 0..31
        float4 v = ((const float4*)(A + (size_t)(row0 + r) * K))[c4];
        *(float4*)&As[r * LDA_P + c4 * 4] = v;
    }
    __syncthreads();

    const int lane = tid & 31;
    const int wave = tid >> 5;
    const int col0 = wave * 16;                  // this wave's N tile

    const int mrow = lane & 15;                  // A-fragment row within 16
    const int kgrp = (lane >> 4) << 1;           // 0 or 2 (K sub-pair)
    const int ccol = lane & 15;                  // B / C column

    const float* As0 = &As[(mrow +  0) * LDA_P];
    const float* As1 = &As[(mrow + 16) * LDA_P];
    const float* As2 = &As[(mrow + 32) * LDA_P];
    const float* As3 = &As[(mrow + 48) * LDA_P];
    const float* Wp  = W + (size_t)(col0 + ccol) * K;

    v8f acc0 = {}, acc1 = {}, acc2 = {}, acc3 = {};

    for (int k0 = 0; k0 < K; k0 += 4) {
        const int kk = k0 + kgrp;
        v2f bf = *(const v2f*)&Wp[kk];           // W[col][kk], W[col][kk+1]
        v2f a0 = *(const v2f*)&As0[kk];
        v2f a1 = *(const v2f*)&As1[kk];
        v2f a2 = *(const v2f*)&As2[kk];
        v2f a3 = *(const v2f*)&As3[kk];

        acc0 = __builtin_amdgcn_wmma_f32_16x16x4_f32(false, a0, false, bf, (short)0, acc0, false, false);
        acc1 = __builtin_amdgcn_wmma_f32_16x16x4_f32(false, a1, false, bf, (short)0, acc1, false, false);
        acc2 = __builtin_amdgcn_wmma_f32_16x16x4_f32(false, a2, false, bf, (short)0, acc2, false, false);
        acc3 = __builtin_amdgcn_wmma_f32_16x16x4_f32(false, a3, false, bf, (short)0, acc3, false, false);
    }

    const float bv = bias ? bias[col0 + ccol] : 0.0f;
    const int rbase = row0 + 8 * (lane >> 4);    // C row base for this lane
    v8f accs[4] = {acc0, acc1, acc2, acc3};
#pragma unroll
    for (int t = 0; t < 4; ++t) {
#pragma unroll
        for (int v = 0; v < 8; ++v) {
            float x = accs[t][v] + bv;
            if (do_relu) x = fmaxf(x, 0.0f);
            C[(size_t)(rbase + t*16 + v) * DD + col0 + ccol] = x;
        }
    }
}

// ---------------------------------------------------------------------------
// Per-(bn,h) attention logits for the multi-head stage: s,t = Wh . a_src/a_dst
// ---------------------------------------------------------------------------
__global__ void head_st_kernel(const float* __restrict__ Wh,
                               const float* __restrict__ a_src,
                               const float* __restrict__ a_dst,
                               float* __restrict__ sArr, float* __restrict__ tArr)
{
    int i  = blockIdx.x * blockDim.x + threadIdx.x;     // (bn,h), 131072 total
    int h  = i & 7;
    int bn = i >> 3;
    const float* p  = Wh + (size_t)bn * DD + h * FF;
    const float* as = a_src + h * FF;
    const float* ad = a_dst + h * FF;
    float ss = 0.f, tt = 0.f;
#pragma unroll
    for (int f = 0; f < FF; ++f) { float w = p[f]; ss += w * as[f]; tt += w * ad[f]; }
    sArr[i] = ss; tArr[i] = tt;
}

// ---------------------------------------------------------------------------
// Multi-head tri-diagonal attention + ELU + per-channel BN stats (sum, sumsq)
// One thread per (bn,h); produces 16 channel values.
// ---------------------------------------------------------------------------
__global__ void __launch_bounds__(256)
head_att_kernel(const float* __restrict__ Wh, const float* __restrict__ sArr,
                const float* __restrict__ tArr, const float* __restrict__ a_b_h,
                float* __restrict__ hp, float* __restrict__ stats)
{
    __shared__ float lsum[DD];
    __shared__ float lsq[DD];
    int tid = threadIdx.x;
    if (tid < DD) { lsum[tid] = 0.f; lsq[tid] = 0.f; }
    __syncthreads();

    int i  = blockIdx.x * 256 + tid;     // (bn,h)
    int h  = i & 7;
    int bn = i >> 3;
    int n  = bn & (GSZ - 1);
    int b  = bn >> 10;
    float bias = a_b_h[h];
    float si   = sArr[i];

    float e[3];
    int   mc[3];
#pragma unroll
    for (int j = 0; j < 3; ++j) {
        int m = n - 1 + j;
        bool valid = (m >= 0) && (m < GSZ);
        int mcl = valid ? m : n;
        mc[j] = mcl;
        if (valid) {
            float v = si + tArr[(size_t)((b << 10) + m) * HH + h] + bias;
            e[j] = (v >= 0.f) ? v : ALPHA * v;    // LeakyReLU
        } else {
            e[j] = NEGV;
        }
    }
    float mx = fmaxf(e[0], fmaxf(e[1], e[2]));
    float w0 = expf(e[0] - mx), w1 = expf(e[1] - mx), w2 = expf(e[2] - mx);
    float inv = 1.0f / (w0 + w1 + w2);
    w0 *= inv; w1 *= inv; w2 *= inv;

    const float* r0 = Wh + (size_t)((b << 10) + mc[0]) * DD + h * FF;
    const float* r1 = Wh + (size_t)((b << 10) + mc[1]) * DD + h * FF;
    const float* r2 = Wh + (size_t)((b << 10) + mc[2]) * DD + h * FF;
    float* out = hp + (size_t)bn * DD + h * FF;
#pragma unroll
    for (int f = 0; f < FF; ++f) {
        float v = w0 * r0[f] + w1 * r1[f] + w2 * r2[f];
        v = (v > 0.f) ? v : expm1f(v);            // ELU
        out[f] = v;
        atomicAdd(&lsum[h * FF + f], v);
        atomicAdd(&lsq[h * FF + f], v * v);
    }
    __syncthreads();
    if (tid < DD) {
        atomicAdd(&stats[tid], lsum[tid]);
        atomicAdd(&stats[DD + tid], lsq[tid]);
    }
}

// ---------------------------------------------------------------------------
// BN finalize: 128 channels -> scale/shift
// ---------------------------------------------------------------------------
__global__ void bn_finalize_kernel(const float* __restrict__ stats,
                                   const float* __restrict__ g,
                                   const float* __restrict__ bb,
                                   float* __restrict__ scale, float* __restrict__ shift)
{
    int c = threadIdx.x;                       // 128 threads
    const float cnt = (float)MTOT;
    float m  = stats[c] / cnt;
    float v  = stats[DD + c] / cnt - m * m;
    float sc = g[c] * rsqrtf(v + BN_EPS);
    scale[c] = sc;
    shift[c] = bb[c] - m * sc;
}

// float4-vectorized BN apply: out = in*scale[c] + shift[c]
__global__ void bn_apply_kernel(const float4* __restrict__ in,
                                const float* __restrict__ scale,
                                const float* __restrict__ shift,
                                float4* __restrict__ out)
{
    size_t i = (size_t)blockIdx.x * blockDim.x + threadIdx.x;  // MTOT*DD/4
    int c0 = (int)((i << 2) & (DD - 1));
    float4 v = in[i];
    float4 r;
    r.x = v.x * scale[c0 + 0] + shift[c0 + 0];
    r.y = v.y * scale[c0 + 1] + shift[c0 + 1];
    r.z = v.z * scale[c0 + 2] + shift[c0 + 2];
    r.w = v.w * scale[c0 + 3] + shift[c0 + 3];
    out[i] = r;
}

// ---------------------------------------------------------------------------
// out_att logits: s2/t2 = Wh2 . a_src_o / a_dst_o  (one thread per row)
// ---------------------------------------------------------------------------
__global__ void st2_kernel(const float* __restrict__ Wh2,
                           const float* __restrict__ a_src,
                           const float* __restrict__ a_dst,
                           float* __restrict__ s2, float* __restrict__ t2)
{
    int bn = blockIdx.x * blockDim.x + threadIdx.x;   // 16384
    const float* p = Wh2 + (size_t)bn * DD;
    float ss = 0.f, tt = 0.f;
    for (int d = 0; d < DD; ++d) { float w = p[d]; ss += w * a_src[d]; tt += w * a_dst[d]; }
    s2[bn] = ss; t2[bn] = tt;
}

__global__ void att2w_kernel(const float* __restrict__ s2, const float* __restrict__ t2,
                             const float* __restrict__ a_b_o, float* __restrict__ aw)
{
    int bn = blockIdx.x * blockDim.x + threadIdx.x;   // 16384
    int n = bn & (GSZ - 1);
    int b = bn >> 10;
    float bias = a_b_o[0];
    float si = s2[bn];
    float e[3];
#pragma unroll
    for (int j = 0; j < 3; ++j) {
        int m = n - 1 + j;
        bool valid = (m >= 0) && (m < GSZ);
        if (valid) {
            float v = si + t2[(b << 10) + m] + bias;
            e[j] = (v >= 0.f) ? v : ALPHA * v;
        } else e[j] = NEGV;
    }
    float mx = fmaxf(e[0], fmaxf(e[1], e[2]));
    float w0 = expf(e[0] - mx), w1 = expf(e[1] - mx), w2 = expf(e[2] - mx);
    float inv = 1.0f / (w0 + w1 + w2);
    aw[(size_t)bn * 3 + 0] = w0 * inv;
    aw[(size_t)bn * 3 + 1] = w1 * inv;
    aw[(size_t)bn * 3 + 2] = w2 * inv;
}

// ---------------------------------------------------------------------------
// hp2 = att2-blend of Wh2 rows; x2pre = elu(hp2) + x; accumulate BN1 stats
// One thread per element.
// ---------------------------------------------------------------------------
__global__ void __launch_bounds__(256)
combine2_kernel(const float* __restrict__ Wh2, const float* __restrict__ aw,
                const float* __restrict__ x, float* __restrict__ x2pre,
                float* __restrict__ stats)
{
    __shared__ float lsum[DD];
    __shared__ float lsq[DD];
    int tid = threadIdx.x;
    if (tid < DD) { lsum[tid] = 0.f; lsq[tid] = 0.f; }
    __syncthreads();

    size_t i = (size_t)blockIdx.x * 256 + tid;   // MTOT*DD elements
    int c  = (int)(i & (DD - 1));
    int bn = (int)(i >> 7);
    int n  = bn & (GSZ - 1);
    int b  = bn >> 10;
    int m0 = (n > 0) ? n - 1 : n;
    int m2 = (n < GSZ - 1) ? n + 1 : n;
    const float* a3 = aw + (size_t)bn * 3;
    float hp2 = a3[0] * Wh2[(size_t)((b << 10) + m0) * DD + c]
              + a3[1] * Wh2[(size_t)bn * DD + c]
              + a3[2] * Wh2[(size_t)((b << 10) + m2) * DD + c];
    float v = (hp2 > 0.f) ? hp2 : expm1f(hp2);
    v += x[i];
    x2pre[i] = v;
    atomicAdd(&lsum[c], v);
    atomicAdd(&lsq[c], v * v);
    __syncthreads();
    if (tid < DD) {
        atomicAdd(&stats[tid], lsum[tid]);
        atomicAdd(&stats[DD + tid], lsq[tid]);
    }
}

// ---------------------------------------------------------------------------
// x3pre = ff + x2 ; accumulate BN2 stats
// ---------------------------------------------------------------------------
__global__ void __launch_bounds__(256)
resid_stats_kernel(const float* __restrict__ ff, const float* __restrict__ x2,
                   float* __restrict__ x3pre, float* __restrict__ stats)
{
    __shared__ float lsum[DD];
    __shared__ float lsq[DD];
    int tid = threadIdx.x;
    if (tid < DD) { lsum[tid] = 0.f; lsq[tid] = 0.f; }
    __syncthreads();

    size_t i = (size_t)blockIdx.x * 256 + tid;
    int c = (int)(i & (DD - 1));
    float v = ff[i] + x2[i];
    x3pre[i] = v;
    atomicAdd(&lsum[c], v);
    atomicAdd(&lsq[c], v * v);
    __syncthreads();
    if (tid < DD) {
        atomicAdd(&stats[tid], lsum[tid]);
        atomicAdd(&stats[DD + tid], lsq[tid]);
    }
}

// ---------------------------------------------------------------------------
extern "C" void kernel_launch(void* const* d_in, const int* in_sizes, int n_in,
                              void* d_out, int out_size, void* d_ws, size_t ws_size,
                              hipStream_t stream)
{
    const float* x       = (const float*)d_in[0];
    const float* W_heads = (const float*)d_in[1];   // [H*F=128, 128]
    const float* a_src_h = (const float*)d_in[2];   // [8,16]
    const float* a_dst_h = (const float*)d_in[3];
    const float* a_b_h   = (const float*)d_in[4];   // [8]
    const float* bn_g_h  = (const float*)d_in[5];   // [128] flat
    const float* bn_b_h  = (const float*)d_in[6];
    const float* W_out   = (const float*)d_in[7];   // [128,128]
    const float* a_src_o = (const float*)d_in[8];   // [128]
    const float* a_dst_o = (const float*)d_in[9];
    const float* a_b_o   = (const float*)d_in[10];  // scalar
    const float* W1      = (const float*)d_in[11];
    const float* b1      = (const float*)d_in[12];
    const float* W2      = (const float*)d_in[13];
    const float* b2      = (const float*)d_in[14];
    const float* bn1_g   = (const float*)d_in[15];
    const float* bn1_b   = (const float*)d_in[16];
    const float* bn2_g   = (const float*)d_in[17];
    const float* bn2_b   = (const float*)d_in[18];
    float* out = (float*)d_out;

    const size_t NEL = (size_t)MTOT * DD;           // 2,097,152
    float* B0    = (float*)d_ws;                    // 8 MB
    float* B1    = B0 + NEL;                        // 8 MB
    float* B2    = B1 + NEL;                        // 8 MB
    float* sH    = B2 + NEL;                        // 131072
    float* tH    = sH + (size_t)MTOT * HH;          // 131072
    float* s2    = tH + (size_t)MTOT * HH;          // 16384
    float* t2    = s2 + MTOT;                       // 16384
    float* aw    = t2 + MTOT;                       // 49152
    float* stats = aw + (size_t)MTOT * 3;           // 3 * 256
    float* scale = stats + 768;                     // 128
    float* shift = scale + DD;                      // 128

    hipMemsetAsync(stats, 0, 768 * sizeof(float), stream);

    const int gemmBlocks = MTOT / 64;               // 256 blocks of 8 waves
    const dim3 B256(256);

    // ---- stage 1: multi-head GAT ----
    gemm_nt_wmma<<<gemmBlocks, B256, 0, stream>>>(x, W_heads, nullptr, B0, MTOT, 0); // B0=Wh
    head_st_kernel<<<(MTOT * HH) / 256, B256, 0, stream>>>(B0, a_src_h, a_dst_h, sH, tH);
    head_att_kernel<<<(MTOT * HH) / 256, B256, 0, stream>>>(B0, sH, tH, a_b_h, B1, stats); // B1=elu(hp)
    bn_finalize_kernel<<<1, 128, 0, stream>>>(stats, bn_g_h, bn_b_h, scale, shift);
    bn_apply_kernel<<<NEL / 4 / 256, B256, 0, stream>>>((const float4*)B1, scale, shift, (float4*)B2); // B2=x1

    // ---- stage 2: out_att ----
    gemm_nt_wmma<<<gemmBlocks, B256, 0, stream>>>(B2, W_out, nullptr, B0, MTOT, 0);  // B0=Wh2
    st2_kernel<<<MTOT / 256, B256, 0, stream>>>(B0, a_src_o, a_dst_o, s2, t2);
    att2w_kernel<<<MTOT / 256, B256, 0, stream>>>(s2, t2, a_b_o, aw);
    combine2_kernel<<<NEL / 256, B256, 0, stream>>>(B0, aw, x, B1, stats + 256);     // B1=x2pre
    bn_finalize_kernel<<<1, 128, 0, stream>>>(stats + 256, bn1_g, bn1_b, scale, shift);
    bn_apply_kernel<<<NEL / 4 / 256, B256, 0, stream>>>((const float4*)B1, scale, shift, (float4*)B2); // B2=x2

    // ---- stage 3: FFN + residual + BN ----
    gemm_nt_wmma<<<gemmBlocks, B256, 0, stream>>>(B2, W1, b1, B0, MTOT, 1);          // B0=relu(x2@W1^T+b1)
    gemm_nt_wmma<<<gemmBlocks, B256, 0, stream>>>(B0, W2, b2, B1, MTOT, 0);          // B1=ff
    resid_stats_kernel<<<NEL / 256, B256, 0, stream>>>(B1, B2, B0, stats + 512);     // B0=ff+x2
    bn_finalize_kernel<<<1, 128, 0, stream>>>(stats + 512, bn2_g, bn2_b, stats ? scale : scale, shift);
    bn_apply_kernel<<<NEL / 4 / 256, B256, 0, stream>>>((const float4*)B0, scale, shift, (float4*)out);
}